// TBRSurface_46626164966024
// MI455X (gfx1250) — compile-verified
//
#include <hip/hip_runtime.h>
#include <stdint.h>

#define W_DIM 304
#define H_DIM 240
#define BINS  8
#define HWSZ  (H_DIM * W_DIM)
#define TILE  256   // events per block (8 wave32s)

typedef __attribute__((address_space(3))) float lds_float;

// ---------------------------------------------------------------------------
// Scatter: stage 256-event tile (5120 B) into LDS via async b128 copies,
// then atomicOr bit (1 << t_bin) into the per-pixel bitmask.
// ---------------------------------------------------------------------------
__global__ void __launch_bounds__(TILE) scatter_kernel(
    const float* __restrict__ ev,
    const float* __restrict__ params,
    unsigned*    __restrict__ mask,
    int n)
{
    __shared__ float tile[TILE * 5];          // 5120 bytes
    const int tileStart = blockIdx.x * TILE;
    const bool full = (tileStart + TILE) <= n;

    float xf, yf, tf, bf;
    if (full) {
        // Low 32 bits of the generic LDS pointer == workgroup-relative offset.
        unsigned ldsBase = (unsigned)(uintptr_t)(&tile[0]);
        // Launder: compiler can no longer trace DMA target / reads to 'tile',
        // so it cannot fold the LDS reads of DMA-written data to undef.
        asm volatile("" : "+v"(ldsBase));

        const float* src = ev + (size_t)tileStart * 5;   // 16B-aligned tile
        for (int k = (int)threadIdx.x; k < (TILE * 5) / 4; k += TILE) {
            unsigned     loff = ldsBase + (unsigned)k * 16u;
            const float* g    = src + (size_t)k * 4;
            asm volatile("global_load_async_to_lds_b128 %0, %1, off"
                         :: "v"(loff), "v"(g) : "memory");
        }
        asm volatile("s_wait_asynccnt 0" ::: "memory");  // my wave's copies done
        __syncthreads();                                  // publish all waves

        // Real ds_load_b32 reads; stride 5 dwords -> conflict-free on 64 banks.
        lds_float* lp = (lds_float*)(ldsBase + (unsigned)threadIdx.x * 20u);
        xf = lp[0];
        yf = lp[1];
        tf = lp[2];
        bf = lp[4];                           // p (index 3) unused by reference
    }

    int e = tileStart + (int)threadIdx.x;
    if (e < n) {
        if (!full) {                          // ragged tail: direct global reads
            size_t o = (size_t)e * 5;
            xf = ev[o + 0]; yf = ev[o + 1]; tf = ev[o + 2]; bf = ev[o + 4];
        }
        int   bi = (int)bf;
        float t0 = params[2 * bi];
        float dn = params[2 * bi + 1];
        float tn = (tf - t0) / dn;
        int bin = (int)(tn * (float)BINS);    // truncation == astype(int32)
        bin = bin < 0 ? 0 : (bin > BINS - 1 ? BINS - 1 : bin);
        int cell = bi * HWSZ + (int)yf * W_DIM + (int)xf;
        atomicOr(&mask[cell], 1u << bin);     // global_atomic_or_b32, no return
    }
}

// ---------------------------------------------------------------------------
// Per-segment t0 and denom = (tN - t0 + 1e-5). B=64 -> trivial.
// ---------------------------------------------------------------------------
__global__ void seg_params_kernel(const float* __restrict__ ev,
                                  const int*   __restrict__ lengths,
                                  float*       __restrict__ params,
                                  int nseg)
{
    int b = blockIdx.x * blockDim.x + threadIdx.x;
    if (b >= nseg) return;
    long start = 0;
#pragma clang loop vectorize(disable) unroll(disable)
    for (int j = 0; j < b; ++j) start += lengths[j];
    long last  = start + lengths[b] - 1;
    float t0 = ev[(size_t)start * 5 + 2];
    float tN = ev[(size_t)last  * 5 + 2];
    params[2 * b]     = t0;
    params[2 * b + 1] = (tN - t0) + 1e-5f;
}

// ---------------------------------------------------------------------------
// Zero the bitmask (aliases d_out), vectorized b128 stores.
// ---------------------------------------------------------------------------
__global__ void zero_kernel(unsigned* __restrict__ p, int n4, int ntot)
{
    int i = blockIdx.x * blockDim.x + threadIdx.x;
    if (i < n4) {
        ((uint4*)p)[i] = make_uint4(0u, 0u, 0u, 0u);
    }
    if (i == 0) {                             // scalar tail (robustness)
        for (int r = n4 * 4; r < ntot; ++r) p[r] = 0u;
    }
}

// ---------------------------------------------------------------------------
// In-place mask(uint32) -> float mask/256: the power-of-2 bin reduction
// Σ bit·2^bin is exactly the mask integer.
// ---------------------------------------------------------------------------
__global__ void finalize_kernel(unsigned* p, int n4, int ntot)
{
    const float s = 1.0f / 256.0f;
    int i = blockIdx.x * blockDim.x + threadIdx.x;
    if (i < n4) {
        uint4 v = ((uint4*)p)[i];             // read own cells, then overwrite
        float4 r;
        r.x = (float)v.x * s; r.y = (float)v.y * s;
        r.z = (float)v.z * s; r.w = (float)v.w * s;
        ((float4*)p)[i] = r;
    }
    if (i == 0) {
        for (int r2 = n4 * 4; r2 < ntot; ++r2) {
            unsigned u = p[r2];
            ((float*)p)[r2] = (float)u * s;
        }
    }
}

// ---------------------------------------------------------------------------
extern "C" void kernel_launch(void* const* d_in, const int* in_sizes, int n_in,
                              void* d_out, int out_size, void* d_ws, size_t ws_size,
                              hipStream_t stream)
{
    const float* ev      = (const float*)d_in[0];   // [N,5] f32
    const int*   lengths = (const int*)  d_in[1];   // [B]   i32
    int n    = in_sizes[0] / 5;                     // events
    int nseg = in_sizes[1];                         // segments

    float*    params = (float*)d_ws;                // 2*nseg floats (512 B)
    unsigned* mask   = (unsigned*)d_out;            // bitmask aliases output

    seg_params_kernel<<<(nseg + 63) / 64, 64, 0, stream>>>(ev, lengths, params, nseg);

    int n4 = out_size / 4;
    zero_kernel<<<(n4 + 255) / 256, 256, 0, stream>>>(mask, n4, out_size);

    scatter_kernel<<<(n + TILE - 1) / TILE, TILE, 0, stream>>>(ev, params, mask, n);

    finalize_kernel<<<(n4 + 255) / 256, 256, 0, stream>>>(mask, n4, out_size);
}